// GAT_11390253269557
// MI455X (gfx1250) — compile-verified
//
#include <hip/hip_runtime.h>

// ---------------- problem constants (from reference) ----------------
#define NNODES   50000
#define NEDGES   800000
#define ETOT     (NEDGES + NNODES)   // edges + self loops
#define NGRAPH   256
#define HC       256                 // HEADS*HID
#define HEADS    4
#define HID      64
#define NEG_SLOPE 0.2f
#define EPS_BN   1e-5f

typedef __attribute__((ext_vector_type(2))) float v2f;
typedef __attribute__((ext_vector_type(8))) float v8f;
typedef __attribute__((ext_vector_type(4))) int   v4i;

// async global->LDS path (gfx1250): guarded so either toolchain still compiles.
// Signature (from clang diagnostic): (AS1 v4i*, AS3 v4i*, imm int, imm int)
#define ASYNC_LDS_OK (__has_builtin(__builtin_amdgcn_global_load_async_to_lds_b128) && \
                      __has_builtin(__builtin_amdgcn_s_wait_asynccnt))
typedef __attribute__((address_space(1))) v4i* g_v4i_ptr;  // pointer to global int4
typedef __attribute__((address_space(3))) v4i* l_v4i_ptr;  // pointer to LDS int4

// order-preserving float<->uint mapping for atomicMax on f32 (handles negatives)
__device__ __forceinline__ unsigned f2ord(float f) {
  unsigned b = __float_as_uint(f);
  return (b & 0x80000000u) ? ~b : (b | 0x80000000u);
}
__device__ __forceinline__ float ord2f(unsigned u) {
  unsigned b = (u & 0x80000000u) ? (u & 0x7FFFFFFFu) : ~u;
  return __uint_as_float(b);
}
__device__ __forceinline__ float lrelu(float v) { return v > 0.f ? v : NEG_SLOPE * v; }

__device__ __forceinline__ void edge_sd(const int* __restrict__ ei, int e, int& s, int& d) {
  if (e < NEDGES) { s = ei[e]; d = ei[NEDGES + e]; }
  else            { s = e - NEDGES; d = s; }          // appended self-loop
}

// ---------------- layer-0 projection: h = x[50000,9] @ W0[9,256] ----------------
__global__ __launch_bounds__(256)
void k_proj9(const float* __restrict__ x, const float* __restrict__ W,
             float* __restrict__ H) {
  __shared__ float xs[9];
  const int n = blockIdx.x;
  const int c = threadIdx.x;
  if (c < 9) xs[c] = x[(size_t)n * 9 + c];
  __syncthreads();
  float s = 0.f;
#pragma unroll
  for (int k = 0; k < 9; ++k) s += xs[k] * W[k * HC + c];
  H[(size_t)n * HC + c] = s;
}

// ---------------- f32 WMMA GEMM: C[M,256] = relu?(A[M,256] @ B[256,256] + bias?) ----------------
// Block = 128 threads (4 wave32). Block owns 16-row M tile (LDS-staged via async engine).
// Each wave covers 64 output columns = 4 WMMA N-tiles.
// V_WMMA_F32_16X16X4_F32 lane layout (ISA 7.12.2):
//   A 16x4: lanes 0-15 row=lane, vgpr{0,1}=K{0,1}; lanes 16-31 vgpr{0,1}=K{2,3}
//   C/D:    lane (g=lane>>4, r=lane&15): vgpr v -> row v+8g, col r
__global__ __launch_bounds__(128)
void k_gemm_wmma(const float* __restrict__ A, const float* __restrict__ B,
                 const float* __restrict__ bias, float* __restrict__ C,
                 int M, int fuse_relu) {
  __shared__ float As[16 * 256];
  const int t  = threadIdx.x;
  const int m0 = blockIdx.x * 16;
  {
    const float4* Ag  = (const float4*)(A + (size_t)m0 * 256);
    float4*       As4 = (float4*)As;
#if ASYNC_LDS_OK
    // global_load_async_to_lds_b128: direct HBM/L2 -> LDS, tracked by ASYNCcnt,
    // no VGPR round-trip for the staging tile.
#pragma unroll
    for (int i = 0; i < 8; ++i) {
      const int idx = t + i * 128;
      __builtin_amdgcn_global_load_async_to_lds_b128(
          (g_v4i_ptr)(Ag + idx), (l_v4i_ptr)(As4 + idx), 0, 0);
    }
    __builtin_amdgcn_s_wait_asynccnt(0);
#else
#pragma unroll
    for (int i = 0; i < 8; ++i) As4[t + i * 128] = Ag[t + i * 128];
#endif
  }
  __syncthreads();
  const int lane = t & 31;
  const int g    = lane >> 4;
  const int r    = lane & 15;
  const int n0   = (t >> 5) * 64;

  v8f acc0 = {}, acc1 = {}, acc2 = {}, acc3 = {};
  for (int k0 = 0; k0 < 256; k0 += 4) {
    const int kk = k0 + 2 * g;
    v2f a = *(const v2f*)(&As[r * 256 + kk]);        // ds_load_b64
    const float* B0 = B + (size_t)kk * 256 + n0 + r; // row K=kk
    const float* B1 = B0 + 256;                      // row K=kk+1
    __builtin_prefetch(B0 + 1024, 0, 1);             // next k-panel -> global_prefetch_b8
    v2f b0 = { B0[0],  B1[0]  };
    v2f b1 = { B0[16], B1[16] };
    v2f b2 = { B0[32], B1[32] };
    v2f b3 = { B0[48], B1[48] };
    acc0 = __builtin_amdgcn_wmma_f32_16x16x4_f32(false, a, false, b0, (short)0, acc0, false, false);
    acc1 = __builtin_amdgcn_wmma_f32_16x16x4_f32(false, a, false, b1, (short)0, acc1, false, false);
    acc2 = __builtin_amdgcn_wmma_f32_16x16x4_f32(false, a, false, b2, (short)0, acc2, false, false);
    acc3 = __builtin_amdgcn_wmma_f32_16x16x4_f32(false, a, false, b3, (short)0, acc3, false, false);
  }
#pragma unroll
  for (int v = 0; v < 8; ++v) {
    const int row = m0 + v + 8 * g;
    if (row < M) {
      float* crow = C + (size_t)row * 256 + n0 + r;
      float o0 = acc0[v], o1 = acc1[v], o2 = acc2[v], o3 = acc3[v];
      if (bias) {
        o0 += bias[n0 + r];      o1 += bias[n0 + 16 + r];
        o2 += bias[n0 + 32 + r]; o3 += bias[n0 + 48 + r];
      }
      if (fuse_relu) {
        o0 = fmaxf(o0, 0.f); o1 = fmaxf(o1, 0.f);
        o2 = fmaxf(o2, 0.f); o3 = fmaxf(o3, 0.f);
      }
      crow[0] = o0; crow[16] = o1; crow[32] = o2; crow[48] = o3;
    }
  }
}

// ---------------- attention logits: al[n,h] = sum_c h[n,h,c]*a[h,c] ----------------
// one wave32 per node; lane covers 8 contiguous channels; head = lane/8
__global__ __launch_bounds__(256)
void k_attn(const float* __restrict__ H, const float* __restrict__ a_src,
            const float* __restrict__ a_dst, float* __restrict__ alsrc,
            float* __restrict__ aldst) {
  const int n = (blockIdx.x * blockDim.x + threadIdx.x) >> 5;
  if (n >= NNODES) return;
  const int lane = threadIdx.x & 31;
  const float4* hr = (const float4*)(H + (size_t)n * HC);
  float4 h0 = hr[lane * 2], h1 = hr[lane * 2 + 1];
  const float4* sr = (const float4*)a_src;
  const float4* dr = (const float4*)a_dst;
  float4 s0 = sr[lane * 2], s1 = sr[lane * 2 + 1];
  float4 d0 = dr[lane * 2], d1 = dr[lane * 2 + 1];
  float ps = h0.x*s0.x + h0.y*s0.y + h0.z*s0.z + h0.w*s0.w
           + h1.x*s1.x + h1.y*s1.y + h1.z*s1.z + h1.w*s1.w;
  float pd = h0.x*d0.x + h0.y*d0.y + h0.z*d0.z + h0.w*d0.w
           + h1.x*d1.x + h1.y*d1.y + h1.z*d1.z + h1.w*d1.w;
#pragma unroll
  for (int off = 1; off < 8; off <<= 1) {           // reduce within each 8-lane head group
    ps += __shfl_xor(ps, off, 32);
    pd += __shfl_xor(pd, off, 32);
  }
  if ((lane & 7) == 0) {
    const int head = lane >> 3;
    alsrc[(size_t)n * HEADS + head] = ps;
    aldst[(size_t)n * HEADS + head] = pd;
  }
}

// ---------------- edge pass 1: segment max of e over dst ----------------
__global__ __launch_bounds__(256)
void k_edge_max(const int* __restrict__ ei, const float* __restrict__ alsrc,
                const float* __restrict__ aldst, unsigned* __restrict__ emax) {
  const int e = blockIdx.x * blockDim.x + threadIdx.x;
  if (e >= ETOT) return;
  int s, d; edge_sd(ei, e, s, d);
  float4 as = *(const float4*)(alsrc + (size_t)s * HEADS);
  float4 ad = *(const float4*)(aldst + (size_t)d * HEADS);
  atomicMax(&emax[(size_t)d * HEADS + 0], f2ord(lrelu(as.x + ad.x)));
  atomicMax(&emax[(size_t)d * HEADS + 1], f2ord(lrelu(as.y + ad.y)));
  atomicMax(&emax[(size_t)d * HEADS + 2], f2ord(lrelu(as.z + ad.z)));
  atomicMax(&emax[(size_t)d * HEADS + 3], f2ord(lrelu(as.w + ad.w)));
}

// ---------------- edge pass 2: denom = segment sum of exp(e - emax[dst]) ----------------
__global__ __launch_bounds__(256)
void k_edge_denom(const int* __restrict__ ei, const float* __restrict__ alsrc,
                  const float* __restrict__ aldst, const unsigned* __restrict__ emax,
                  float* __restrict__ denom) {
  const int e = blockIdx.x * blockDim.x + threadIdx.x;
  if (e >= ETOT) return;
  int s, d; edge_sd(ei, e, s, d);
  float4 as = *(const float4*)(alsrc + (size_t)s * HEADS);
  float4 ad = *(const float4*)(aldst + (size_t)d * HEADS);
  const float m0 = ord2f(emax[(size_t)d * HEADS + 0]);
  const float m1 = ord2f(emax[(size_t)d * HEADS + 1]);
  const float m2 = ord2f(emax[(size_t)d * HEADS + 2]);
  const float m3 = ord2f(emax[(size_t)d * HEADS + 3]);
  atomicAdd(&denom[(size_t)d * HEADS + 0], expf(lrelu(as.x + ad.x) - m0));
  atomicAdd(&denom[(size_t)d * HEADS + 1], expf(lrelu(as.y + ad.y) - m1));
  atomicAdd(&denom[(size_t)d * HEADS + 2], expf(lrelu(as.z + ad.z) - m2));
  atomicAdd(&denom[(size_t)d * HEADS + 3], expf(lrelu(as.w + ad.w) - m3));
}

// ---------------- edge pass 3: out[dst] += alpha * h[src] ----------------
// one wave32 per edge; lane handles 8 channels; head = lane/8
__global__ __launch_bounds__(256)
void k_edge_accum(const int* __restrict__ ei, const float* __restrict__ alsrc,
                  const float* __restrict__ aldst, const unsigned* __restrict__ emax,
                  const float* __restrict__ denom, const float* __restrict__ H,
                  float* __restrict__ out) {
  const int e = (blockIdx.x * blockDim.x + threadIdx.x) >> 5;
  if (e >= ETOT) return;
  const int lane = threadIdx.x & 31;
  int s, d; edge_sd(ei, e, s, d);
  const int head = lane >> 3;
  const float v = lrelu(alsrc[(size_t)s * HEADS + head] + aldst[(size_t)d * HEADS + head]);
  const float alpha = expf(v - ord2f(emax[(size_t)d * HEADS + head])) /
                      (denom[(size_t)d * HEADS + head] + 1e-16f);
  const float4* hp = (const float4*)(H + (size_t)s * HC + lane * 8);
  float4 p0 = hp[0], p1 = hp[1];
  float* op = out + (size_t)d * HC + lane * 8;
  atomicAdd(op + 0, alpha * p0.x); atomicAdd(op + 1, alpha * p0.y);
  atomicAdd(op + 2, alpha * p0.z); atomicAdd(op + 3, alpha * p0.w);
  atomicAdd(op + 4, alpha * p1.x); atomicAdd(op + 5, alpha * p1.y);
  atomicAdd(op + 6, alpha * p1.z); atomicAdd(op + 7, alpha * p1.w);
}

// ---------------- batchnorm stats: per-channel sum / sumsq ----------------
#define BN_ROWS 128
__global__ __launch_bounds__(256)
void k_bn_stats(const float* __restrict__ X, float* __restrict__ sums) {
  const int c = threadIdx.x;
  const int r0 = blockIdx.x * BN_ROWS;
  const int r1 = min(r0 + BN_ROWS, NNODES);
  float s = 0.f, s2 = 0.f;
  for (int n = r0; n < r1; ++n) {
    const float v = X[(size_t)n * HC + c];   // coalesced across threads
    s += v; s2 += v * v;
  }
  atomicAdd(&sums[c], s);
  atomicAdd(&sums[HC + c], s2);
}

// ---------------- batchnorm apply + relu ----------------
__global__ __launch_bounds__(256)
void k_bn_apply(const float* __restrict__ X, const float* __restrict__ sums,
                const float* __restrict__ gamma, const float* __restrict__ beta,
                float* __restrict__ Y) {
  const size_t i = (size_t)blockIdx.x * blockDim.x + threadIdx.x;
  if (i >= (size_t)NNODES * HC) return;
  const int c = (int)(i & (HC - 1));
  const float inv_n = 1.0f / (float)NNODES;
  const float mu  = sums[c] * inv_n;
  const float var = sums[HC + c] * inv_n - mu * mu;
  const float v = (X[i] - mu) * rsqrtf(var + EPS_BN) * gamma[c] + beta[c];
  Y[i] = v > 0.f ? v : 0.f;
}

// ---------------- row softmax + per-graph scatter: f[batch[n]] += softmax(x[n]) ----------------
__global__ __launch_bounds__(256)
void k_softmax_scatter(const float* __restrict__ X, const int* __restrict__ batch,
                       float* __restrict__ F) {
  const int n = (blockIdx.x * blockDim.x + threadIdx.x) >> 5;
  if (n >= NNODES) return;
  const int lane = threadIdx.x & 31;
  const float4* xr = (const float4*)(X + (size_t)n * HC);
  float4 a = xr[lane * 2], b = xr[lane * 2 + 1];
  float m = fmaxf(fmaxf(fmaxf(a.x, a.y), fmaxf(a.z, a.w)),
                  fmaxf(fmaxf(b.x, b.y), fmaxf(b.z, b.w)));
#pragma unroll
  for (int off = 1; off < 32; off <<= 1) m = fmaxf(m, __shfl_xor(m, off, 32));
  float e0 = expf(a.x - m), e1 = expf(a.y - m), e2 = expf(a.z - m), e3 = expf(a.w - m);
  float e4 = expf(b.x - m), e5 = expf(b.y - m), e6 = expf(b.z - m), e7 = expf(b.w - m);
  float s = e0 + e1 + e2 + e3 + e4 + e5 + e6 + e7;
#pragma unroll
  for (int off = 1; off < 32; off <<= 1) s += __shfl_xor(s, off, 32);
  const float inv = 1.0f / s;
  float* fp = F + (size_t)batch[n] * HC + lane * 8;
  atomicAdd(fp + 0, e0 * inv); atomicAdd(fp + 1, e1 * inv);
  atomicAdd(fp + 2, e2 * inv); atomicAdd(fp + 3, e3 * inv);
  atomicAdd(fp + 4, e4 * inv); atomicAdd(fp + 5, e5 * inv);
  atomicAdd(fp + 6, e6 * inv); atomicAdd(fp + 7, e7 * inv);
}

// ---------------- host orchestration ----------------
extern "C" void kernel_launch(void* const* d_in, const int* in_sizes, int n_in,
                              void* d_out, int out_size, void* d_ws, size_t ws_size,
                              hipStream_t stream) {
  (void)in_sizes; (void)n_in; (void)out_size; (void)ws_size;
  const float* x     = (const float*)d_in[0];
  const int*   ei    = (const int*)  d_in[1];
  const int*   batch = (const int*)  d_in[2];
  const float* W0    = (const float*)d_in[3];
  const float* as0   = (const float*)d_in[4];
  const float* ad0   = (const float*)d_in[5];
  // d_in[6] = bias0, d_in[10] = bias12: constant-per-channel bias cancels in batchnorm
  const float* W12   = (const float*)d_in[7];
  const float* as12  = (const float*)d_in[8];
  const float* ad12  = (const float*)d_in[9];
  const float* gamma = (const float*)d_in[11];
  const float* beta  = (const float*)d_in[12];
  const float* Wm1   = (const float*)d_in[13];
  const float* bm1   = (const float*)d_in[14];
  const float* Wm2   = (const float*)d_in[15];
  const float* bm2   = (const float*)d_in[16];

  // workspace carve-out (256B aligned)
  char* p = (char*)d_ws;
  auto take = [&](size_t bytes) -> char* {
    char* r = p; p += (bytes + 255) & ~(size_t)255; return r;
  };
  float*    bufA  = (float*)   take((size_t)NNODES * HC * 4);  // 51.2 MB
  float*    bufB  = (float*)   take((size_t)NNODES * HC * 4);  // 51.2 MB
  float*    alsrc = (float*)   take((size_t)NNODES * HEADS * 4);
  float*    aldst = (float*)   take((size_t)NNODES * HEADS * 4);
  unsigned* emax  = (unsigned*)take((size_t)NNODES * HEADS * 4);
  float*    denom = (float*)   take((size_t)NNODES * HEADS * 4);
  float*    bnsum = (float*)   take(2 * HC * 4);
  float*    fbuf  = (float*)   take(NGRAPH * HC * 4);
  float*    t1    = (float*)   take(NGRAPH * HC * 4);

  const int GRID_EDGE  = (ETOT + 255) / 256;          // thread-per-edge kernels
  const int GRID_EWAVE = (ETOT * 32 + 255) / 256;     // wave-per-edge kernel
  const int GRID_NWAVE = (NNODES * 32 + 255) / 256;   // wave-per-node kernels
  const int GRID_BN    = (NNODES + BN_ROWS - 1) / BN_ROWS;
  const int GRID_APPLY = (int)(((size_t)NNODES * HC + 255) / 256);

  (void)hipMemsetAsync(fbuf, 0, (size_t)NGRAPH * HC * 4, stream);

  const float* cur = nullptr;  // current node features (HC cols), ws-resident
  for (int layer = 0; layer < 3; ++layer) {
    float* hb = (layer & 1) ? bufB : bufA;  // h = x @ W
    float* ag = (layer & 1) ? bufA : bufB;  // aggregation target

    // 1) dense projection
    if (layer == 0) {
      k_proj9<<<NNODES, 256, 0, stream>>>(x, W0, hb);
    } else {
      k_gemm_wmma<<<NNODES / 16, 128, 0, stream>>>(
          cur, W12 + (size_t)(layer - 1) * HC * HC, nullptr, hb, NNODES, 0);
    }

    // 2) attention logits
    const float* asrc = (layer == 0) ? as0 : as12 + (size_t)(layer - 1) * HEADS * HID;
    const float* adst = (layer == 0) ? ad0 : ad12 + (size_t)(layer - 1) * HEADS * HID;
    k_attn<<<GRID_NWAVE, 256, 0, stream>>>(hb, asrc, adst, alsrc, aldst);

    // 3) zero segment buffers (mapped-uint 0 == "-inf" sentinel for emax)
    (void)hipMemsetAsync(emax,  0, (size_t)NNODES * HEADS * 4, stream);
    (void)hipMemsetAsync(denom, 0, (size_t)NNODES * HEADS * 4, stream);
    (void)hipMemsetAsync(ag,    0, (size_t)NNODES * HC * 4, stream);
    (void)hipMemsetAsync(bnsum, 0, 2 * HC * 4, stream);

    // 4) edge-softmax + weighted aggregation
    k_edge_max  <<<GRID_EDGE,  256, 0, stream>>>(ei, alsrc, aldst, emax);
    k_edge_denom<<<GRID_EDGE,  256, 0, stream>>>(ei, alsrc, aldst, emax, denom);
    k_edge_accum<<<GRID_EWAVE, 256, 0, stream>>>(ei, alsrc, aldst, emax, denom, hb, ag);

    // 5) batchnorm + relu (writes next-layer features over hb)
    k_bn_stats<<<GRID_BN,    256, 0, stream>>>(ag, bnsum);
    k_bn_apply<<<GRID_APPLY, 256, 0, stream>>>(ag, bnsum, gamma + layer * HC,
                                               beta + layer * HC, hb);
    cur = hb;

    // 6) row softmax + per-graph accumulation into f
    k_softmax_scatter<<<GRID_NWAVE, 256, 0, stream>>>(cur, batch, fbuf);
  }

  // readout MLP: relu(f @ Wm1 + bm1) @ Wm2 + bm2
  k_gemm_wmma<<<NGRAPH / 16, 128, 0, stream>>>(fbuf, Wm1, bm1, t1, NGRAPH, 1);
  k_gemm_wmma<<<NGRAPH / 16, 128, 0, stream>>>(t1, Wm2, bm2, (float*)d_out, NGRAPH, 0);
}